// MyResidualVQ_45148696216467
// MI455X (gfx1250) — compile-verified
//
#include <hip/hip_runtime.h>
#include <hip/hip_bf16.h>

// Residual VQ for MI455X (gfx1250).
// Score GEMM on fp32 matrix cores: v_wmma_f32_16x16x4_f32, 2Mx2N register
// tiling (4 accumulator chains), shared 32x512 residual tile in LDS,
// codes split across 2 waves, shfl + LDS argmin merge.

typedef __attribute__((ext_vector_type(2))) float v2f;
typedef __attribute__((ext_vector_type(8))) float v8f;

#define RVQ_B 8
#define RVQ_S 4096
#define RVQ_D 512
#define RVQ_Q 8
#define RVQ_C 1024
#define RVQ_T (RVQ_B * RVQ_S)          // 32768 tokens
#define LDS_STRIDE 516                 // 512 + 4 pad -> conflict-free A reads
#define TILE_FLOATS (32 * LDS_STRIDE)  // 32-row residual tile
#define ARGMIN_SHMEM ((TILE_FLOATS + 64 + 64) * 4)

// ---------------------------------------------------------------------------
__global__ __launch_bounds__(32) void rvq_init_losses(float* losses) {
    if (threadIdx.x < RVQ_Q) losses[threadIdx.x] = 0.0f;
}

// ---------------------------------------------------------------------------
// cnorm[q*C + c] = sum_d codebooks[q][c][d]^2   (one block per code row)
__global__ __launch_bounds__(128) void rvq_cnorm_kernel(
    const float* __restrict__ cbs, float* __restrict__ cnorm) {
    __shared__ float red[128];
    const int code = blockIdx.x;               // 0 .. Q*C-1
    const int t    = threadIdx.x;
    const float4* row = (const float4*)(cbs + (size_t)code * RVQ_D);
    float4 v = row[t];                         // 128 threads * 4 floats = 512
    red[t] = v.x * v.x + v.y * v.y + v.z * v.z + v.w * v.w;
    __syncthreads();
    for (int off = 64; off > 0; off >>= 1) {
        if (t < off) red[t] += red[t + off];
        __syncthreads();
    }
    if (t == 0) cnorm[code] = red[0];
}

// ---------------------------------------------------------------------------
// Per-level argmin. Block = 2 waves, 32 tokens. Wave w scans codes
// [w*512, w*512+512). dist/2 = 0.5*||c||^2 - <r, c>;  r = x - quant_so_far.
__global__ __launch_bounds__(64) void rvq_argmin_kernel(
    const float* __restrict__ x,       // [T, D]
    const float* __restrict__ quant,   // [T, D] running quantized sum (d_out)
    const float* __restrict__ cb,      // [C, D] this level's codebook
    const float* __restrict__ cnorm,   // [C]
    int*         __restrict__ idx_out, // [T]   (workspace, int)
    float*       __restrict__ idxf_out,// [T*Q] (d_out indices, as float)
    int level) {
    extern __shared__ float smem[];
    float* tile = smem;                        // [32][LDS_STRIDE]
    float* mkey = smem + TILE_FLOATS;          // [2][32] per-wave best key
    int*   midx = (int*)(mkey + 64);           // [2][32] per-wave best index

    const int tid  = threadIdx.x;              // 0..63
    const int lane = tid & 31;
    const int wave = tid >> 5;
    const int tok0 = blockIdx.x * 32;

    // Cooperative float4 staging of the 32x512 residual tile.
    for (int i = 0; i < 64; ++i) {
        int f  = i * 64 + tid;                 // 0..4095 (float4 index)
        int r  = f >> 7;                       // row 0..31
        int c4 = f & 127;                      // float4 column
        float4 v = ((const float4*)(x + (size_t)(tok0 + r) * RVQ_D))[c4];
        if (level > 0) {
            float4 qv = ((const float4*)(quant + (size_t)(tok0 + r) * RVQ_D))[c4];
            v.x -= qv.x; v.y -= qv.y; v.z -= qv.z; v.w -= qv.w;
        }
        *(float4*)(tile + r * LDS_STRIDE + c4 * 4) = v;
    }
    __syncthreads();

    const int m  = lane & 15;                  // A row / B column for this lane
    const int kh = lane >> 4;                  // K half (ISA 32-bit A/B layout)
    const int cbase = wave * (RVQ_C / 2);      // this wave's code range

    float best0[8], best1[8];
    int   bidx0[8], bidx1[8];
#pragma unroll
    for (int i = 0; i < 8; ++i) {
        best0[i] = 3.4e38f; best1[i] = 3.4e38f;
        bidx0[i] = 0;       bidx1[i] = 0;
    }

    const float* a0p = tile + m * LDS_STRIDE + 2 * kh;        // M rows 0..15
    const float* a1p = a0p + 16 * LDS_STRIDE;                 // M rows 16..31

    for (int nt = 0; nt < (RVQ_C / 2) / 32; ++nt) {           // 16 iters
        const int code0 = cbase + nt * 32 + m;
        const int code1 = code0 + 16;
        const float* b0p = cb + (size_t)code0 * RVQ_D + 2 * kh;
        const float* b1p = cb + (size_t)code1 * RVQ_D + 2 * kh;

        v8f acc00 = {0.f,0.f,0.f,0.f,0.f,0.f,0.f,0.f};
        v8f acc01 = {0.f,0.f,0.f,0.f,0.f,0.f,0.f,0.f};
        v8f acc10 = {0.f,0.f,0.f,0.f,0.f,0.f,0.f,0.f};
        v8f acc11 = {0.f,0.f,0.f,0.f,0.f,0.f,0.f,0.f};
#pragma unroll 4
        for (int k = 0; k < RVQ_D; k += 4) {
            v2f a0 = *(const v2f*)(a0p + k);
            v2f a1 = *(const v2f*)(a1p + k);
            v2f b0 = *(const v2f*)(b0p + k);
            v2f b1 = *(const v2f*)(b1p + k);
            acc00 = __builtin_amdgcn_wmma_f32_16x16x4_f32(
                false, a0, false, b0, (short)0, acc00, false, false);
            acc01 = __builtin_amdgcn_wmma_f32_16x16x4_f32(
                false, a0, false, b1, (short)0, acc01, false, false);
            acc10 = __builtin_amdgcn_wmma_f32_16x16x4_f32(
                false, a1, false, b0, (short)0, acc10, false, false);
            acc11 = __builtin_amdgcn_wmma_f32_16x16x4_f32(
                false, a1, false, b1, (short)0, acc11, false, false);
        }

        const float k0 = 0.5f * cnorm[code0];
        const float k1 = 0.5f * cnorm[code1];
#pragma unroll
        for (int i = 0; i < 8; ++i) {          // row M = 8*kh + i per tile
            float ka = k0 - acc00[i];
            if (ka < best0[i]) { best0[i] = ka; bidx0[i] = code0; }
            float kb = k1 - acc01[i];
            if (kb < best0[i]) { best0[i] = kb; bidx0[i] = code1; }
            float kc = k0 - acc10[i];
            if (kc < best1[i]) { best1[i] = kc; bidx1[i] = code0; }
            float kd = k1 - acc11[i];
            if (kd < best1[i]) { best1[i] = kd; bidx1[i] = code1; }
        }
    }

    // Min-reduce across the 16 lanes of each half-wave (columns of C tile),
    // tie-breaking toward the lower code index.
#pragma unroll
    for (int off = 8; off >= 1; off >>= 1) {
#pragma unroll
        for (int i = 0; i < 8; ++i) {
            float ov = __shfl_xor(best0[i], off, 16);
            int   oi = __shfl_xor(bidx0[i], off, 16);
            if (ov < best0[i] || (ov == best0[i] && oi < bidx0[i])) {
                best0[i] = ov; bidx0[i] = oi;
            }
            float pv = __shfl_xor(best1[i], off, 16);
            int   pi = __shfl_xor(bidx1[i], off, 16);
            if (pv < best1[i] || (pv == best1[i] && pi < bidx1[i])) {
                best1[i] = pv; bidx1[i] = pi;
            }
        }
    }

    if ((lane & 15) == 0) {                    // lane 0 / lane 16 writers
#pragma unroll
        for (int i = 0; i < 8; ++i) {
            int r = 8 * kh + i;
            mkey[wave * 32 + r]      = best0[i];
            midx[wave * 32 + r]      = bidx0[i];
            mkey[wave * 32 + 16 + r] = best1[i];
            midx[wave * 32 + 16 + r] = bidx1[i];
        }
    }
    __syncthreads();

    // Merge the two waves' candidates; one thread per token row.
    if (tid < 32) {
        float v0 = mkey[tid];      int i0 = midx[tid];       // wave0 (lower ids)
        float v1 = mkey[32 + tid]; int i1 = midx[32 + tid];  // wave1
        int fi = (v1 < v0 || (v1 == v0 && i1 < i0)) ? i1 : i0;
        int tok = tok0 + tid;
        idx_out[tok] = fi;
        idxf_out[(size_t)tok * RVQ_Q + level] = (float)fi;
    }
}

// ---------------------------------------------------------------------------
// Per-level update: quant += cb[idx]; loss_q += mean(new_residual^2).
__global__ __launch_bounds__(128) void rvq_update_kernel(
    const float* __restrict__ x,
    const float* __restrict__ cb,
    const int*   __restrict__ idx,
    float*       __restrict__ quant,
    float*       __restrict__ loss,   // &losses[level]
    int level) {
    __shared__ float red[128];
    const int tok = blockIdx.x;
    const int t   = threadIdx.x;

    const float4* crow = (const float4*)(cb + (size_t)idx[tok] * RVQ_D);
    const float4* xrow = (const float4*)(x + (size_t)tok * RVQ_D);
    float4*       qrow = (float4*)(quant + (size_t)tok * RVQ_D);

    float4 cv = crow[t];
    float4 xv = xrow[t];
    float4 qv = (level > 0) ? qrow[t] : make_float4(0.f, 0.f, 0.f, 0.f);

    float rx = (xv.x - qv.x) - cv.x;
    float ry = (xv.y - qv.y) - cv.y;
    float rz = (xv.z - qv.z) - cv.z;
    float rw = (xv.w - qv.w) - cv.w;

    qrow[t] = make_float4(qv.x + cv.x, qv.y + cv.y, qv.z + cv.z, qv.w + cv.w);

    red[t] = rx * rx + ry * ry + rz * rz + rw * rw;
    __syncthreads();
    for (int off = 64; off > 0; off >>= 1) {
        if (t < off) red[t] += red[t + off];
        __syncthreads();
    }
    if (t == 0)
        atomicAdd(loss, red[0] * (1.0f / ((float)RVQ_T * (float)RVQ_D)));
}

// ---------------------------------------------------------------------------
extern "C" void kernel_launch(void* const* d_in, const int* in_sizes, int n_in,
                              void* d_out, int out_size, void* d_ws, size_t ws_size,
                              hipStream_t stream) {
    const float* x   = (const float*)d_in[0];           // [8,4096,512]
    const float* cbs = (const float*)d_in[1];           // [8,1024,512]

    float* out    = (float*)d_out;
    float* quant  = out;                                // [T, D]
    float* idxf   = out + (size_t)RVQ_T * RVQ_D;        // [T, Q]
    float* losses = idxf + (size_t)RVQ_T * RVQ_Q;       // [Q]

    int*   idx_ws = (int*)d_ws;                         // [T]
    float* cnorm  = (float*)d_ws + RVQ_T;               // [Q*C]

    rvq_init_losses<<<1, 32, 0, stream>>>(losses);
    rvq_cnorm_kernel<<<RVQ_Q * RVQ_C, 128, 0, stream>>>(cbs, cnorm);

    for (int q = 0; q < RVQ_Q; ++q) {
        const float* cb = cbs + (size_t)q * RVQ_C * RVQ_D;
        rvq_argmin_kernel<<<RVQ_T / 32, 64, ARGMIN_SHMEM, stream>>>(
            x, quant, cb, cnorm + (size_t)q * RVQ_C, idx_ws, idxf, q);
        rvq_update_kernel<<<RVQ_T, 128, 0, stream>>>(
            x, cb, idx_ws, quant, losses + q, q);
    }
}